// SuperPixelLoss_14053132992787
// MI455X (gfx1250) — compile-verified
//
#include <hip/hip_runtime.h>
#include <hip/hip_bf16.h>

typedef float v2f __attribute__((ext_vector_type(2)));
typedef float v4f __attribute__((ext_vector_type(4)));
typedef float v8f __attribute__((ext_vector_type(8)));

// ---------------------------------------------------------------------------
// Problem constants (from reference): B=8, image 3x1024x1024, feature
// 128x128x128, sp 256x256 with 256 labels, pooled/resized size S=256.
// ---------------------------------------------------------------------------

// Kernel A: stream image (100.7 MB): |x| -> 4x4 avgpool -> channel sum ->
// per-block LDS segment histogram -> global atomic add. Also accumulates
// label counts. One block = one pooled row (256 pixels) of one batch.
__global__ void k_image_seg(const float* __restrict__ img,
                            const int* __restrict__ sp,
                            float* __restrict__ sums1,
                            float* __restrict__ cnts) {
    __shared__ float accv[256];
    __shared__ float accc[256];
    const int t = threadIdx.x;
    accv[t] = 0.f;
    accc[t] = 0.f;
    __syncthreads();

    const int blk = blockIdx.x;        // 0..2047
    const int b   = blk >> 8;          // batch
    const int sy  = blk & 255;         // pooled row
    const int sx  = t;                 // pooled col

    // image flat index: ((b*3 + c)*1024 + (4*sy + r))*1024 + 4*sx
    const float* base = img + ((size_t)(b * 3) * 1024 + (size_t)(4 * sy)) * 1024 + 4 * sx;
    float val = 0.f;
#pragma unroll
    for (int c = 0; c < 3; ++c) {
#pragma unroll
        for (int r = 0; r < 4; ++r) {
            const v4f* q = (const v4f*)(base + (size_t)c * 1024 * 1024 + (size_t)r * 1024);
            v4f v = __builtin_nontemporal_load(q);
            val += fabsf(v.x) + fabsf(v.y) + fabsf(v.z) + fabsf(v.w);
        }
    }
    val *= (1.0f / 16.0f);  // avgpool mean; channel sum already folded in

    const int pix = (b << 16) + (sy << 8) + sx;
    const int label = sp[pix];
    atomicAdd(&accv[label], val);
    atomicAdd(&accc[label], 1.0f);
    __syncthreads();
    atomicAdd(&sums1[(b << 8) + t], accv[t]);
    atomicAdd(&cnts[(b << 8) + t], accc[t]);
}

// Kernel B: stream feature (67.1 MB): reduce over 128 channels ->
// fsum[8,128,128]. Channel-sum commutes with (linear) bilinear resize, so
// this collapses the resize workload by 128x. Each thread owns a float4 of x
// and a 32-channel chunk; chunks combine via global f32 atomics (fsum is
// zero-initialized). 131072 threads -> 4096 waves for latency hiding.
__global__ void k_feat_csum(const float* __restrict__ feat,
                            float* __restrict__ fsum) {
    const int idx   = blockIdx.x * blockDim.x + threadIdx.x;  // 0..131071
    const int chunk = idx >> 15;                              // 0..3 (32 ch each)
    const int rem   = idx & 32767;
    const int b     = rem >> 12;                              // 4096 quads/batch
    const int r2    = rem & 4095;
    const int y     = r2 >> 5;
    const int xq    = r2 & 31;

    const float* base = feat + (size_t)b * (128 * 128 * 128)
                        + (size_t)(chunk * 32) * 16384 + y * 128 + xq * 4;
    v4f acc = {0.f, 0.f, 0.f, 0.f};
#pragma unroll 4
    for (int c = 0; c < 32; ++c) {
        v4f v = __builtin_nontemporal_load((const v4f*)(base + (size_t)c * 16384));
        acc += v;
    }
    float* out = fsum + (size_t)b * 16384 + y * 128 + xq * 4;
    atomicAdd(&out[0], acc.x);
    atomicAdd(&out[1], acc.y);
    atomicAdd(&out[2], acc.z);
    atomicAdd(&out[3], acc.w);
}

// Align-corners interp coefficients for output index s in [0,256) over a
// source of length 128: ys = s*127/255.
__device__ __forceinline__ void interp_coef(int s, int& i0, int& i1, float& w1, float& w0) {
    float ys = (float)(s * 127) / 255.0f;
    i0 = (int)ys;
    i1 = (i0 + 1 < 127) ? (i0 + 1) : 127;
    w1 = ys - (float)i0;
    w0 = 1.0f - w1;
}

// Kernel C1 (WMMA): T[b] (256x128) = Ry (256x128) @ fsum[b] (128x128).
// Ry rows are generated in-register (2 nonzeros per row). One wave per 16x16
// tile; V_WMMA_F32_16X16X4_F32, K stepped by 4.
// f32 A-frag layout (ISA 7.12.2): lanes 0-15 -> K=k0,k0+1 ; lanes 16-31 -> K=k0+2,k0+3.
__global__ void __launch_bounds__(32) k_resize_g1(const float* __restrict__ fsum,
                                                  float* __restrict__ T) {
    const int wid = blockIdx.x;     // 0..1023 (128 tiles/batch)
    const int L   = threadIdx.x;    // 0..31, EXEC all ones (WMMA requirement)
    const int b   = wid >> 7;
    const int t   = wid & 127;
    const int m0  = (t >> 3) * 16;  // output row tile (16 of them)
    const int n0  = (t & 7) * 16;   // output col tile (8 of them)
    const int l15 = L & 15;
    const int hi  = L >> 4;

    // Per-lane A row = m0 + l15 of Ry
    int i0, i1;
    float w1, w0;
    interp_coef(m0 + l15, i0, i1, w1, w0);

    const float* fb = fsum + (size_t)b * 16384;
    v8f c = {};
    for (int k0 = 0; k0 < 128; k0 += 4) {
        const int ka = k0 + hi * 2;
        v2f a, bb;
        a.x = ((ka == i0) ? w0 : 0.f) + ((ka == i1) ? w1 : 0.f);
        a.y = ((ka + 1 == i0) ? w0 : 0.f) + ((ka + 1 == i1) ? w1 : 0.f);
        bb.x = fb[ka * 128 + n0 + l15];
        bb.y = fb[(ka + 1) * 128 + n0 + l15];
        c = __builtin_amdgcn_wmma_f32_16x16x4_f32(false, a, false, bb, (short)0, c,
                                                  false, false);
    }
    float* Tb = T + (size_t)b * 256 * 128;
#pragma unroll
    for (int r = 0; r < 8; ++r) {
        Tb[(m0 + r + hi * 8) * 128 + n0 + l15] = c[r];
    }
}

// Kernel C2 (WMMA): f256[b] (256x256) = T[b] (256x128) @ RxT (128x256).
// RxT columns generated in-register.
__global__ void __launch_bounds__(32) k_resize_g2(const float* __restrict__ T,
                                                  float* __restrict__ f256) {
    const int wid = blockIdx.x;     // 0..2047 (256 tiles/batch)
    const int L   = threadIdx.x;
    const int b   = wid >> 8;
    const int t   = wid & 255;
    const int m0  = (t >> 4) * 16;
    const int n0  = (t & 15) * 16;
    const int l15 = L & 15;
    const int hi  = L >> 4;

    // Per-lane B column = n0 + l15 of RxT
    int i0, i1;
    float w1, w0;
    interp_coef(n0 + l15, i0, i1, w1, w0);

    const int arow = m0 + l15;
    const float* Tb = T + (size_t)b * 256 * 128;
    v8f c = {};
    for (int k0 = 0; k0 < 128; k0 += 4) {
        const int ka = k0 + hi * 2;
        v2f a, bb;
        a.x = Tb[arow * 128 + ka];
        a.y = Tb[arow * 128 + ka + 1];
        bb.x = ((ka == i0) ? w0 : 0.f) + ((ka == i1) ? w1 : 0.f);
        bb.y = ((ka + 1 == i0) ? w0 : 0.f) + ((ka + 1 == i1) ? w1 : 0.f);
        c = __builtin_amdgcn_wmma_f32_16x16x4_f32(false, a, false, bb, (short)0, c,
                                                  false, false);
    }
    float* ob = f256 + (size_t)b * 65536;
#pragma unroll
    for (int r = 0; r < 8; ++r) {
        ob[(m0 + r + hi * 8) * 256 + n0 + l15] = c[r];
    }
}

// Kernel D: segment-sum the resized channel-summed feature.
__global__ void k_feat_seg(const float* __restrict__ f256,
                           const int* __restrict__ sp,
                           float* __restrict__ sums2) {
    __shared__ float accv[256];
    const int t = threadIdx.x;
    accv[t] = 0.f;
    __syncthreads();
    const int blk = blockIdx.x;  // 0..2047
    const int b   = blk >> 8;
    const int sy  = blk & 255;
    const int pix = (b << 16) + (sy << 8) + t;
    const float val = f256[pix];
    const int label = sp[pix];
    atomicAdd(&accv[label], val);
    __syncthreads();
    atomicAdd(&sums2[(b << 8) + t], accv[t]);
}

// Kernel E: means + pairwise similarity-difference reduction.
// loss = (1/(B*N*N)) * sum_b sum_ij | |m1_i-m1_j| - |m2_i-m2_j| |
__global__ void k_final(const float* __restrict__ sums1,
                        const float* __restrict__ cnts,
                        const float* __restrict__ sums2,
                        float* __restrict__ out) {
    __shared__ float m1s[256];
    __shared__ float m2s[256];
    __shared__ float red[256];
    const int b = blockIdx.x;
    const int t = threadIdx.x;
    const float c  = cnts[(b << 8) + t];
    const float m1 = (c > 0.f) ? sums1[(b << 8) + t] / (c * 3.0f) : 0.f;
    const float m2 = (c > 0.f) ? sums2[(b << 8) + t] / (c * 128.0f) : 0.f;
    m1s[t] = m1;
    m2s[t] = m2;
    __syncthreads();
    float acc = 0.f;
    for (int j = 0; j < 256; ++j) {
        acc += fabsf(fabsf(m1 - m1s[j]) - fabsf(m2 - m2s[j]));
    }
    red[t] = acc;
    __syncthreads();
    for (int s = 128; s > 0; s >>= 1) {
        if (t < s) red[t] += red[t + s];
        __syncthreads();
    }
    if (t == 0) atomicAdd(out, red[0] * (1.0f / (8.0f * 256.0f * 256.0f)));
}

extern "C" void kernel_launch(void* const* d_in, const int* in_sizes, int n_in,
                              void* d_out, int out_size, void* d_ws, size_t ws_size,
                              hipStream_t stream) {
    const float* image   = (const float*)d_in[0];  // [8,3,1024,1024] f32
    const float* feature = (const float*)d_in[1];  // [8,128,128,128] f32
    const int*   sp      = (const int*)d_in[2];    // [8,1,256,256] i32
    // d_in[3] = num (==256), constant-folded.

    float* ws    = (float*)d_ws;
    float* sums1 = ws;                  //   2048 f
    float* cnts  = ws + 2048;           //   2048 f
    float* sums2 = ws + 4096;           //   2048 f
    float* fsum  = ws + 6144;           // 131072 f  [8,128,128]
    float* Tbuf  = fsum + 131072;       // 262144 f  [8,256,128]
    float* f256  = Tbuf + 262144;       // 524288 f  [8,256,256]

    // Zero accumulators (sums1/cnts/sums2/fsum) and the scalar output.
    hipMemsetAsync(ws, 0, (size_t)(6144 + 131072) * sizeof(float), stream);
    hipMemsetAsync(d_out, 0, sizeof(float), stream);

    k_image_seg<<<2048, 256, 0, stream>>>(image, sp, sums1, cnts);
    k_feat_csum<<<512, 256, 0, stream>>>(feature, fsum);
    k_resize_g1<<<1024, 32, 0, stream>>>(fsum, Tbuf);
    k_resize_g2<<<2048, 32, 0, stream>>>(Tbuf, f256);
    k_feat_seg<<<2048, 256, 0, stream>>>(f256, sp, sums2);
    k_final<<<8, 256, 0, stream>>>(sums1, cnts, sums2, (float*)d_out);
}